// EventTape_369367187857
// MI455X (gfx1250) — compile-verified
//
#include <hip/hip_runtime.h>
#include <math.h>

typedef __attribute__((ext_vector_type(2))) float v2f;
typedef __attribute__((ext_vector_type(8))) float v8f;

#define B_DIM   8
#define T_LEN   4096
#define D_DIM   1024
#define DENT    256
#define ENTRY   1280
#define EMAX    32
#define TVOCAB  512
#define KC      128
#define NTHR    256

// ---------------------------------------------------------------------------
// Kernel 1: event selection (threshold / top-k fallback / ascending-time sort)
// One block per batch. Writes float valid/times into d_out tail, int copies
// into d_ws for the GEMM kernel.
// ---------------------------------------------------------------------------
__global__ __launch_bounds__(NTHR)
void select_events_kernel(const float* __restrict__ z,
                          float* __restrict__ out_valid_f,
                          float* __restrict__ out_times_f,
                          int* __restrict__ ws_times,
                          int* __restrict__ ws_valid) {
    __shared__ float s_sur[T_LEN];
    __shared__ float s_score[T_LEN];
    __shared__ float red_val[NTHR];
    __shared__ int   red_idx[NTHR];
    __shared__ int   s_top_idx[EMAX];
    __shared__ int   s_top_val[EMAX];
    __shared__ int   s_cnt;

    const int b   = blockIdx.x;
    const int tid = threadIdx.x;
    if (tid == 0) s_cnt = 0;
    __syncthreads();

    // surprise = max_j |z[b,t,j]|, count threshold hits
    const float* zb = z + (size_t)b * T_LEN * 8;
    int local_cnt = 0;
#pragma unroll
    for (int i = 0; i < T_LEN / NTHR; ++i) {
        int t = tid + i * NTHR;
        const float* zp = zb + (size_t)t * 8;
        float m = 0.0f;
#pragma unroll
        for (int j = 0; j < 8; ++j) m = fmaxf(m, fabsf(zp[j]));
        s_sur[t] = m;
        local_cnt += (m > 2.0f) ? 1 : 0;
    }
    atomicAdd(&s_cnt, local_cnt);
    __syncthreads();

    const bool use_fb = (s_cnt < 4);   // top_k fallback if fewer than MIN_EVENTS
#pragma unroll
    for (int i = 0; i < T_LEN / NTHR; ++i) {
        int t = tid + i * NTHR;
        float m = s_sur[t];
        s_score[t] = (use_fb || m > 2.0f) ? m : -__builtin_inff();
    }
    __syncthreads();

    // 32 rounds of argmax (value desc, index asc on ties) == lax.top_k order.
    // Selected entries are marked NaN (never re-selected: all compares fail).
    for (int r = 0; r < EMAX; ++r) {
        float bv = -__builtin_inff();
        int   bi = T_LEN;
#pragma unroll
        for (int i = 0; i < T_LEN / NTHR; ++i) {
            int t = tid + i * NTHR;
            float v = s_score[t];
            if (v > bv || (v == bv && t < bi)) { bv = v; bi = t; }
        }
        red_val[tid] = bv;
        red_idx[tid] = bi;
        __syncthreads();
        for (int s = NTHR / 2; s > 0; s >>= 1) {
            if (tid < s) {
                float ov = red_val[tid + s];
                int   oi = red_idx[tid + s];
                if (ov > red_val[tid] || (ov == red_val[tid] && oi < red_idx[tid])) {
                    red_val[tid] = ov;
                    red_idx[tid] = oi;
                }
            }
            __syncthreads();
        }
        if (tid == 0) {
            int idx = red_idx[0];
            s_top_idx[r] = idx;
            s_top_val[r] = (use_fb || (s_sur[idx] > 2.0f)) ? 1 : 0;
            s_score[idx] = __builtin_nanf("");
        }
        __syncthreads();
    }

    // Stable placement: valid entries sorted ascending by time at the front,
    // invalid slots pushed to the end with times=0, valid=0.
    if (tid < EMAX) {
        int idx = s_top_idx[tid];
        int val = s_top_val[tid];
        int nvalid = 0, pos = 0;
#pragma unroll
        for (int q = 0; q < EMAX; ++q) {
            nvalid += s_top_val[q];
            pos += (s_top_val[q] && s_top_idx[q] < idx) ? 1 : 0;
        }
        int invpos = 0;
        for (int q = 0; q < tid; ++q) invpos += (s_top_val[q] == 0) ? 1 : 0;
        int slot = val ? pos : (nvalid + invpos);

        out_times_f[b * EMAX + slot] = val ? (float)idx : 0.0f;
        out_valid_f[b * EMAX + slot] = val ? 1.0f : 0.0f;
        ws_times[b * EMAX + slot]    = val ? idx : 0;
        ws_valid[b * EMAX + slot]    = val;
    }
}

// ---------------------------------------------------------------------------
// Kernel 2: gather + GEMM via V_WMMA_F32_16X16X4_F32.
//   out[m, d] = sum_f raw[m, f] * W[d, f] + b_proj[d] + time_embed[min(t,511), d]
//   raw[m, :1024] = valid * h_seq[b, t, :],  raw[m, 1024:] = valid * ent_flat[b, t, :]
// Grid: 16 blocks (16 event rows each), 256 threads = 8 waves,
// each wave owns 128 output columns (8 x 16x16 accumulator tiles).
// ---------------------------------------------------------------------------
__global__ __launch_bounds__(NTHR)
void event_tape_gemm_kernel(const float* __restrict__ h_seq,
                            const float* __restrict__ ent,     // (B,T,256) flat
                            const float* __restrict__ W,       // (1024,1280)
                            const float* __restrict__ b_proj,  // (1024)
                            const float* __restrict__ t_emb,   // (512,1024)
                            const int* __restrict__ ws_times,
                            const int* __restrict__ ws_valid,
                            float* __restrict__ out) {
    // A staging: 16 rows x KC floats, row stride padded to 132 dwords
    // (132 mod 64 = 4 -> 16 rows land on distinct LDS bank groups).
    __shared__ float sA[16][KC + 4];
    __shared__ int   s_t[16];
    __shared__ int   s_tclip[16];
    __shared__ float s_vm[16];

    const int mb  = blockIdx.x;       // 16 event rows: m = mb*16 + r
    const int tid = threadIdx.x;
    const int lane = tid & 31;
    const int wave = tid >> 5;

    if (tid < 16) {
        int m = mb * 16 + tid;
        int t = ws_times[m];
        s_t[tid]     = t;
        s_tclip[tid] = (t < TVOCAB - 1) ? t : (TVOCAB - 1);
        s_vm[tid]    = ws_valid[m] ? 1.0f : 0.0f;
    }
    __syncthreads();

    const int col0 = wave * 128;          // this wave's 128 output columns
    v8f acc[8] = {};                      // 8 x (16x16 f32) accumulator tiles

    // A-fragment addressing per ISA 16x4 f32 layout:
    //   lanes 0-15: row M = lane, K offsets {0,1}; lanes 16-31: row M = lane-16, K {2,3}
    const int arow  = lane & 15;
    const int akoff = (lane >> 4) << 1;   // 0 or 2

    for (int k0 = 0; k0 < ENTRY; k0 += KC) {
        __syncthreads();   // protect sA reads of previous chunk

        // Stage masked raw rows: 16 rows x 128 floats, 8 floats per thread.
        {
            int r   = tid >> 4;            // row 0..15
            int off = (tid & 15) * 8;      // 0..120
            int f   = k0 + off;
            int m   = mb * 16 + r;
            int bb  = m >> 5;              // batch
            int t   = s_t[r];
            float vm = s_vm[r];
            const float* src = (f < D_DIM)
                ? (h_seq + ((size_t)bb * T_LEN + t) * D_DIM + f)
                : (ent   + ((size_t)bb * T_LEN + t) * DENT + (f - D_DIM));
            float4 p0 = *(const float4*)(src);
            float4 p1 = *(const float4*)(src + 4);
            p0.x *= vm; p0.y *= vm; p0.z *= vm; p0.w *= vm;
            p1.x *= vm; p1.y *= vm; p1.z *= vm; p1.w *= vm;
            *(float4*)&sA[r][off]     = p0;
            *(float4*)&sA[r][off + 4] = p1;
        }
        __syncthreads();

        // MAC: K in steps of 4, 8 column tiles per wave.
        for (int kk = 0; kk < KC; kk += 4) {
            v2f av = *(const v2f*)&sA[arow][kk + akoff];
            int bk = k0 + kk + akoff;
#pragma unroll
            for (int j = 0; j < 8; ++j) {
                // B per ISA 4x16 layout: lane = column d, K pair {akoff, akoff+1}
                int d = col0 + j * 16 + (lane & 15);
                v2f bv = *(const v2f*)(W + (size_t)d * ENTRY + bk);
                acc[j] = __builtin_amdgcn_wmma_f32_16x16x4_f32(
                    /*neg_a=*/false, av, /*neg_b=*/false, bv,
                    /*c_mod=*/(short)0, acc[j],
                    /*reuse_a=*/false, /*reuse_b=*/false);
            }
        }
    }

    // Epilogue: C/D layout -> VGPR r holds row M=r (lanes 0-15) / M=8+r (16-31),
    // column N = lane&15 within each 16-wide tile.
    const int rbase = (lane >> 4) << 3;   // 0 or 8
#pragma unroll
    for (int j = 0; j < 8; ++j) {
        int d = col0 + j * 16 + (lane & 15);
        float bp = b_proj[d];
#pragma unroll
        for (int r = 0; r < 8; ++r) {
            int mr = rbase + r;                     // row within 16-row tile
            int m  = mb * 16 + mr;                  // global event row (b*32+e)
            float te = t_emb[(size_t)s_tclip[mr] * D_DIM + d];
            out[(size_t)m * D_DIM + d] = acc[j][r] + bp + te;
        }
    }
}

// ---------------------------------------------------------------------------
// Launch
// ---------------------------------------------------------------------------
extern "C" void kernel_launch(void* const* d_in, const int* in_sizes, int n_in,
                              void* d_out, int out_size, void* d_ws, size_t ws_size,
                              hipStream_t stream) {
    const float* h_seq   = (const float*)d_in[0];   // (8,4096,1024)
    const float* z       = (const float*)d_in[1];   // (8,4096,8)
    const float* ent     = (const float*)d_in[2];   // (8,4096,8,32) -> (8,4096,256)
    const float* W_proj  = (const float*)d_in[3];   // (1024,1280)
    const float* b_proj  = (const float*)d_in[4];   // (1024)
    const float* t_emb   = (const float*)d_in[5];   // (512,1024)

    float* out_entries = (float*)d_out;                       // 8*32*1024
    float* out_valid   = out_entries + B_DIM * EMAX * D_DIM;  // 256 floats
    float* out_times   = out_valid + B_DIM * EMAX;            // 256 floats

    int* ws_times = (int*)d_ws;                     // 256 ints
    int* ws_valid = ws_times + B_DIM * EMAX;        // 256 ints

    select_events_kernel<<<B_DIM, NTHR, 0, stream>>>(
        z, out_valid, out_times, ws_times, ws_valid);

    event_tape_gemm_kernel<<<(B_DIM * EMAX) / 16, NTHR, 0, stream>>>(
        h_seq, ent, W_proj, b_proj, t_emb, ws_times, ws_valid, out_entries);
}